// Transformer_56599079027080
// MI455X (gfx1250) — compile-verified
//
#include <hip/hip_runtime.h>

// ---------------------------------------------------------------------------
// Transformer block forward for MI455X (gfx1250, wave32, WMMA).
// GEMMs: batched WMMA-f32 kernel (v_wmma_f32_16x16x4_f32, exact fp32) with
// double-buffered GLOBAL_LOAD_ASYNC_TO_LDS_B128 staging (ASYNCcnt pipeline).
// ---------------------------------------------------------------------------

#define BM 128
#define BN 128
#define BK 16
#define LDP 18          // padded K-stride (floats) for K-contiguous LDS tiles
#define LN_EPS 1e-5f

typedef float v2f __attribute__((ext_vector_type(2)));
typedef float v8f __attribute__((ext_vector_type(8)));

// CDNA5 async copy: 16B global -> 16B LDS, tracked by ASYNCcnt (ISA 15.18.3
// opcode 98: VDST = LDS byte address VGPR, VADDR = 64-bit global address).
__device__ __forceinline__ void async_copy_b128(void* lds_dst, const void* gsrc)
{
    unsigned int       l = (unsigned int)(uintptr_t)lds_dst;   // low 32 = LDS addr
    unsigned long long g = (unsigned long long)(uintptr_t)gsrc;
    asm volatile("global_load_async_to_lds_b128 %0, %1, off"
                 :: "v"(l), "v"(g) : "memory");
}

template <int N>
__device__ __forceinline__ void wait_asynccnt()
{
    asm volatile("s_wait_asynccnt %0" :: "n"(N) : "memory");
}

// ---------------------------------------------------------------------------
// Embedding + positional encoding + the reference's (seq, feature) triangular
// mask:  xm[b,s,d] = (d > s) ? 0 : (emb[tok] + pe[s]).  res keeps unmasked x.
// ---------------------------------------------------------------------------
__global__ __launch_bounds__(256) void embed_kernel(
    const int* __restrict__ tokens, const float* __restrict__ emb,
    const float* __restrict__ pe, float* __restrict__ resx,
    float* __restrict__ xm)
{
    int row = blockIdx.x;          // b*S + s, S = 512
    int s   = row & 511;
    int tok = tokens[row];
    const float* e = emb + (size_t)tok * 512;
    const float* p = pe  + (size_t)s   * 512;
    for (int c = threadIdx.x; c < 512; c += 256) {
        float v = e[c] + p[c];
        resx[(size_t)row * 512 + c] = v;
        xm  [(size_t)row * 512 + c] = (c <= s) ? v : 0.0f;
    }
}

// ---------------------------------------------------------------------------
// Batched GEMM:  C = alpha * A * op(B) + bias  (+ optional ReLU)
//   A: MxK row-major (lda). TRANSB=0: B is KxN row-major (ldb).
//   TRANSB=1: B is NxK row-major (ldb), C[m,n] = sum_k A[m,k]*B[n,k].
// Per-z offsets: off = (z % zdiv)*s?0 + (z / zdiv)*s?1  (floats).
// 256 threads = 8 waves (4x2); wave tile 32x64 = 2x4 WMMA accumulators.
// K staged BK=16 at a time via double-buffered async-to-LDS copies.
// All shapes divide the 128x128 tile exactly — no bounds checks.
// ---------------------------------------------------------------------------
template <int TRANSB>
__global__ __launch_bounds__(256) void gemm_wmma(
    const float* __restrict__ A, const float* __restrict__ B,
    float* __restrict__ C, const float* __restrict__ bias,
    int M, int N, int K, int lda, int ldb, int ldc,
    long sA0, long sA1, long sB0, long sB1, long sC0, long sC1,
    long sBias0, long sBias1, int zdiv, float alpha, int relu)
{
    int z  = blockIdx.z;
    int zi = z % zdiv, zo = z / zdiv;
    const float* Ab = A + (size_t)zi * sA0 + (size_t)zo * sA1;
    const float* Bb = B + (size_t)zi * sB0 + (size_t)zo * sB1;
    float*       Cb = C + (size_t)zi * sC0 + (size_t)zo * sC1;
    const float* biasb = bias ? (bias + (size_t)zi * sBias0 + (size_t)zo * sBias1)
                              : nullptr;

    int m0  = blockIdx.y * BM;
    int n0  = blockIdx.x * BN;
    int tid = threadIdx.x;
    int lane = tid & 31;
    int w    = tid >> 5;
    int wm   = w >> 1;        // 0..3  (M row of 32)
    int wn   = w & 1;         // 0..1  (N col of 64)
    int ml   = lane & 15;     // M (A-frag) / N (B-frag) index within 16
    int kh   = lane >> 4;     // lanes >=16 carry the K+2..K+3 half

    // K-contiguous tiles (async b128 chunks land contiguously; padded stride
    // LDP=18 so 8-byte fragment loads don't all hit the same banks).
    __shared__ float laS[2][BM * LDP];                              // [m][k]
    __shared__ float lbS[2][TRANSB ? (BN * LDP) : (BK * BN)];       // see below

    v8f acc[2][4];
    #pragma unroll
    for (int i = 0; i < 2; ++i)
        #pragma unroll
        for (int j = 0; j < 4; ++j)
            #pragma unroll
            for (int r = 0; r < 8; ++r)
                acc[i][j][r] = 0.0f;

    // Issue one BK stage of async copies into buffer `buf` (4 ops/thread).
    auto issue_stage = [&](int buf, int k0) {
        #pragma unroll
        for (int t = 0; t < 2; ++t) {           // A: 128x16 = 512 b128 chunks
            int c  = tid + t * 256;
            int m  = c >> 2;
            int q4 = (c & 3) << 2;
            async_copy_b128(&laS[buf][m * LDP + q4],
                            Ab + (size_t)(m0 + m) * lda + k0 + q4);
        }
        if (TRANSB) {                           // B NxK: rows K-contiguous
            #pragma unroll
            for (int t = 0; t < 2; ++t) {
                int c  = tid + t * 256;
                int n  = c >> 2;
                int q4 = (c & 3) << 2;
                async_copy_b128(&lbS[buf][n * LDP + q4],
                                Bb + (size_t)(n0 + n) * ldb + k0 + q4);
            }
        } else {                                // B KxN: rows N-contiguous
            #pragma unroll
            for (int t = 0; t < 2; ++t) {
                int c  = tid + t * 256;
                int k  = c >> 5;
                int n4 = (c & 31) << 2;
                async_copy_b128(&lbS[buf][k * BN + n4],
                                Bb + (size_t)(k0 + k) * ldb + n0 + n4);
            }
        }
    };

    int nstage = K / BK;
    issue_stage(0, 0);
    for (int i = 0; i < nstage; ++i) {
        int buf = i & 1;
        if (i + 1 < nstage) {
            issue_stage(buf ^ 1, (i + 1) * BK);
            wait_asynccnt<4>();   // 8 in flight; <=4 left == stage i complete
        } else {
            wait_asynccnt<0>();
        }
        __syncthreads();          // all waves' async data visible

        #pragma unroll
        for (int kk = 0; kk < 4; ++kk) {
            int kb = (kk << 2) + (kh << 1);     // this lane-half's K pair
            v2f af[2];
            #pragma unroll
            for (int mi = 0; mi < 2; ++mi)
                af[mi] = *(const v2f*)&laS[buf][(wm * 32 + mi * 16 + ml) * LDP + kb];
            v2f bfg[4];
            #pragma unroll
            for (int nj = 0; nj < 4; ++nj) {
                int col = wn * 64 + nj * 16 + ml;
                if (TRANSB) {
                    bfg[nj] = *(const v2f*)&lbS[buf][col * LDP + kb];
                } else {
                    v2f t = { lbS[buf][kb * BN + col], lbS[buf][(kb + 1) * BN + col] };
                    bfg[nj] = t;
                }
            }
            #pragma unroll
            for (int mi = 0; mi < 2; ++mi)
                #pragma unroll
                for (int nj = 0; nj < 4; ++nj)
                    acc[mi][nj] = __builtin_amdgcn_wmma_f32_16x16x4_f32(
                        false, af[mi], false, bfg[nj], (short)0,
                        acc[mi][nj], false, false);
        }
        __syncthreads();          // done reading buf before it is re-filled
    }

    // Epilogue. C/D layout: VGPR r -> M = r + 8*(lane>=16), N = lane%16.
    #pragma unroll
    for (int mi = 0; mi < 2; ++mi)
        #pragma unroll
        for (int nj = 0; nj < 4; ++nj)
            #pragma unroll
            for (int r = 0; r < 8; ++r) {
                int m = m0 + wm * 32 + mi * 16 + r + kh * 8;
                int n = n0 + wn * 64 + nj * 16 + ml;
                float v = acc[mi][nj][r] * alpha;
                if (biasb) v += biasb[n];
                if (relu)  v = fmaxf(v, 0.0f);
                Cb[(size_t)m * ldc + n] = v;
            }
}

// ---------------------------------------------------------------------------
// Row softmax over 512 columns, one block (256 thr, 2 elems/thr) per row.
// ---------------------------------------------------------------------------
__global__ __launch_bounds__(256) void softmax_kernel(float* __restrict__ x)
{
    size_t row = blockIdx.x;
    float* p = x + row * 512;
    int tid = threadIdx.x;
    __shared__ float red[256];

    float v0 = p[tid], v1 = p[tid + 256];
    red[tid] = fmaxf(v0, v1);
    __syncthreads();
    for (int s = 128; s > 0; s >>= 1) {
        if (tid < s) red[tid] = fmaxf(red[tid], red[tid + s]);
        __syncthreads();
    }
    float mx = red[0];
    __syncthreads();
    float e0 = __expf(v0 - mx), e1 = __expf(v1 - mx);
    red[tid] = e0 + e1;
    __syncthreads();
    for (int s = 128; s > 0; s >>= 1) {
        if (tid < s) red[tid] += red[tid + s];
        __syncthreads();
    }
    float inv = 1.0f / red[0];
    p[tid] = e0 * inv;
    p[tid + 256] = e1 * inv;
}

// ---------------------------------------------------------------------------
// out = LayerNorm(res + y) * g + b, one block per 512-wide row.
// ---------------------------------------------------------------------------
__global__ __launch_bounds__(256) void add_ln_kernel(
    const float* __restrict__ res, const float* __restrict__ y,
    const float* __restrict__ g, const float* __restrict__ b,
    float* __restrict__ out)
{
    size_t row = blockIdx.x;
    int tid = threadIdx.x;
    __shared__ float red[256];

    float x0 = res[row * 512 + tid]       + y[row * 512 + tid];
    float x1 = res[row * 512 + 256 + tid] + y[row * 512 + 256 + tid];

    red[tid] = x0 + x1;
    __syncthreads();
    for (int s = 128; s > 0; s >>= 1) {
        if (tid < s) red[tid] += red[tid + s];
        __syncthreads();
    }
    float mu = red[0] * (1.0f / 512.0f);
    __syncthreads();

    float d0 = x0 - mu, d1 = x1 - mu;
    red[tid] = d0 * d0 + d1 * d1;
    __syncthreads();
    for (int s = 128; s > 0; s >>= 1) {
        if (tid < s) red[tid] += red[tid + s];
        __syncthreads();
    }
    float rinv = rsqrtf(red[0] * (1.0f / 512.0f) + LN_EPS);

    out[row * 512 + tid]       = g[tid]       * d0 * rinv + b[tid];
    out[row * 512 + 256 + tid] = g[tid + 256] * d1 * rinv + b[tid + 256];
}

// ---------------------------------------------------------------------------
// Driver: B=8, S=512, D=512, H=8, DFF=2048, V=32000.
// ---------------------------------------------------------------------------
extern "C" void kernel_launch(void* const* d_in, const int* in_sizes, int n_in,
                              void* d_out, int out_size, void* d_ws, size_t ws_size,
                              hipStream_t stream)
{
    (void)in_sizes; (void)n_in; (void)out_size; (void)ws_size;

    const int*   tokens = (const int*)  d_in[0];
    const float* emb = (const float*)d_in[1];
    const float* pe  = (const float*)d_in[2];
    const float* Wq  = (const float*)d_in[3];
    const float* bq  = (const float*)d_in[4];
    const float* Wk  = (const float*)d_in[5];
    const float* bk  = (const float*)d_in[6];
    const float* Wv  = (const float*)d_in[7];
    const float* bv  = (const float*)d_in[8];
    const float* Wo  = (const float*)d_in[9];
    const float* bo  = (const float*)d_in[10];
    const float* g1  = (const float*)d_in[11];
    const float* be1 = (const float*)d_in[12];
    const float* W1  = (const float*)d_in[13];
    const float* b1  = (const float*)d_in[14];
    const float* W2  = (const float*)d_in[15];
    const float* b2  = (const float*)d_in[16];
    const float* g2  = (const float*)d_in[17];
    const float* be2 = (const float*)d_in[18];
    const float* Wf  = (const float*)d_in[19];
    const float* bf  = (const float*)d_in[20];
    float* out = (float*)d_out;

    // Workspace carve (floats). Total ~105M floats (~420 MB).
    float* ws = (float*)d_ws;
    size_t off = 0;
    auto carve = [&](size_t n) { float* p = ws + off; off += n; return p; };
    const size_t RD = 4096ull * 512;              // (B*S) x D
    float* xm     = carve(RD);                    // masked input
    float* resx   = carve(RD);                    // residual
    float* qb     = carve(8 * RD);                // [H][B*S][D]
    float* kb     = carve(8 * RD);
    float* vb     = carve(8 * RD);
    float* scores = carve(64ull * 512 * 512);     // [H][B][S][S]
    float* cat    = carve(4096ull * 4096);        // [B*S][H*D]
    float* proj   = carve(RD);
    float* x1     = carve(RD);
    float* hidden = carve(4096ull * 2048);
    float* ff     = carve(RD);
    float* x2     = carve(RD);

    dim3 blk(256);
    const long SD  = 512l * 512;      // per-(h or b) matrix of q/k/v, scores
    const long HSD = 8l * SD;         // per-h stride across all b (q/k/v flat)
    const float scl = 0.044194173824159216f;   // 1/sqrt(512)

    embed_kernel<<<dim3(4096), blk, 0, stream>>>(tokens, emb, pe, resx, xm);

    // Q/K/V projections: per head, C[h] = xm @ Wq[h] + bq[h]   (batch z = h)
    gemm_wmma<0><<<dim3(4, 32, 8), blk, 0, stream>>>(xm, Wq, qb, bq,
        4096, 512, 512, 512, 512, 512,
        0, 0, SD, 0, (long)RD, 0, 512, 0, 8, 1.0f, 0);
    gemm_wmma<0><<<dim3(4, 32, 8), blk, 0, stream>>>(xm, Wk, kb, bk,
        4096, 512, 512, 512, 512, 512,
        0, 0, SD, 0, (long)RD, 0, 512, 0, 8, 1.0f, 0);
    gemm_wmma<0><<<dim3(4, 32, 8), blk, 0, stream>>>(xm, Wv, vb, bv,
        4096, 512, 512, 512, 512, 512,
        0, 0, SD, 0, (long)RD, 0, 512, 0, 8, 1.0f, 0);

    // scores[h,b] = (q[h,b] @ k[h,b]^T) * scale      (z = h*8 + b, zdiv=8)
    gemm_wmma<1><<<dim3(4, 4, 64), blk, 0, stream>>>(qb, kb, scores, nullptr,
        512, 512, 512, 512, 512, 512,
        SD, HSD, SD, HSD, SD, HSD, 0, 0, 8, scl, 0);

    softmax_kernel<<<dim3(64 * 512), blk, 0, stream>>>(scores);

    // o[h,b] = attn @ v[h,b], written straight into cat[b, s, h*D + e]
    gemm_wmma<0><<<dim3(4, 4, 64), blk, 0, stream>>>(scores, vb, cat, nullptr,
        512, 512, 512, 512, 512, 4096,
        SD, HSD, SD, HSD, 512l * 4096, 512, 0, 0, 8, 1.0f, 0);

    // proj = cat @ Wo + bo
    gemm_wmma<0><<<dim3(4, 32, 1), blk, 0, stream>>>(cat, Wo, proj, bo,
        4096, 512, 4096, 4096, 512, 512,
        0, 0, 0, 0, 0, 0, 0, 0, 1, 1.0f, 0);

    add_ln_kernel<<<dim3(4096), blk, 0, stream>>>(resx, proj, g1, be1, x1);

    // FFN
    gemm_wmma<0><<<dim3(16, 32, 1), blk, 0, stream>>>(x1, W1, hidden, b1,
        4096, 2048, 512, 512, 2048, 2048,
        0, 0, 0, 0, 0, 0, 0, 0, 1, 1.0f, 1);
    gemm_wmma<0><<<dim3(4, 32, 1), blk, 0, stream>>>(hidden, W2, ff, b2,
        4096, 512, 2048, 2048, 512, 512,
        0, 0, 0, 0, 0, 0, 0, 0, 1, 1.0f, 0);

    add_ln_kernel<<<dim3(4096), blk, 0, stream>>>(x1, ff, g2, be2, x2);

    // Final vocab projection (dominant GEMM): out = x2 @ Wf + bf
    gemm_wmma<0><<<dim3(250, 32, 1), blk, 0, stream>>>(x2, Wf, out, bf,
        4096, 32000, 512, 512, 32000, 32000,
        0, 0, 0, 0, 0, 0, 0, 0, 1, 1.0f, 0);
}